// AMICOSolver_41953240547521
// MI455X (gfx1250) — compile-verified
//
#include <hip/hip_runtime.h>

// ---------------------------------------------------------------------------
// AMICO ADMM on MI455X (gfx1250, wave32, WMMA)
//   x_{t+1} = W (AtY + rho(z-u)) = C0 + rho * W (z-u),  W = (AtA + rho I)^-1
//   rho = 1.0, kappa = lambda/rho = 0.1, 100 iterations.
// Main cost: 100 x [256x256]x[256x8192] fp32 GEMM -> V_WMMA_F32_16X16X4_F32.
// L2-resident working set (~43 MB << 192 MB) => compute bound, fp32 matrix path.
// ---------------------------------------------------------------------------

#define MM   256      // measurements
#define KK   256      // atoms
#define NN   8192     // voxels
#define RHO_C    1.0f
#define KAPPA_C  0.1f
#define NITER    100

typedef float v2f __attribute__((ext_vector_type(2)));
typedef float v8f __attribute__((ext_vector_type(8)));

// ---------------------------------------------------------------------------
// 1) Aug = [ AtA + rho*I | I ]   (256 x 512)
// ---------------------------------------------------------------------------
__global__ __launch_bounds__(256) void k_gram(const float* __restrict__ A,
                                              float* __restrict__ Aug) {
    int idx = blockIdx.x * 256 + threadIdx.x;      // 256*512 elements
    int r = idx >> 9;
    int c = idx & 511;
    float v;
    if (c < KK) {
        float s = 0.0f;
        #pragma unroll 4
        for (int m = 0; m < MM; ++m)
            s = fmaf(A[m * KK + r], A[m * KK + c], s);
        if (r == c) s += RHO_C;
        v = s;
    } else {
        v = (c - KK == r) ? 1.0f : 0.0f;
    }
    Aug[idx] = v;
}

// ---------------------------------------------------------------------------
// 2) Gauss-Jordan inversion of SPD matrix (no pivoting needed).
//    Single workgroup (1024 threads), pivot row/col staged in LDS so the
//    elimination pass has no cross-thread hazards. Writes W = right half.
// ---------------------------------------------------------------------------
__global__ __launch_bounds__(1024) void k_invert(float* __restrict__ Aug,
                                                 float* __restrict__ Wm) {
    const int tid = threadIdx.x;
    __shared__ float prow[512];
    __shared__ float pcol[256];

    for (int k = 0; k < KK; ++k) {
        float pinv = 1.0f / Aug[k * 512 + k];
        for (int c = tid; c < 512; c += 1024) prow[c] = Aug[k * 512 + c] * pinv;
        if (tid < 256) pcol[tid] = Aug[tid * 512 + k];
        __syncthreads();
        #pragma unroll 4
        for (int e = tid; e < 256 * 512; e += 1024) {
            int r = e >> 9;
            int c = e & 511;
            float v = (r == k) ? prow[c] : fmaf(-pcol[r], prow[c], Aug[e]);
            Aug[e] = v;
        }
        __threadfence();
        __syncthreads();
    }
    for (int e = tid; e < KK * KK; e += 1024)
        Wm[e] = Aug[(e >> 8) * 512 + 256 + (e & 255)];
}

// ---------------------------------------------------------------------------
// 3) AtY[k][n] = sum_m A[m][k] * data[n][m]   (A column broadcast via SGPRs,
//    coalesced stores along n)
// ---------------------------------------------------------------------------
__global__ __launch_bounds__(256) void k_atY(const float* __restrict__ A,
                                             const float* __restrict__ data,
                                             float* __restrict__ AtY) {
    int n = blockIdx.x * 256 + threadIdx.x;  // gridDim.x = 32
    int k = blockIdx.y;                      // gridDim.y = 256
    float s = 0.0f;
    #pragma unroll 4
    for (int m = 0; m < MM; ++m)
        s = fmaf(A[m * KK + k], data[n * MM + m], s);
    AtY[k * NN + n] = s;
}

// ---------------------------------------------------------------------------
// 4) Out[256 x 8192] = Wm[256 x 256] @ B[256 x 8192] (+ Add)
//    Block: 256 threads = 8 waves; block covers full M=256, N tile = 64.
//    Wave tile: 32(M) x 64(N) = 2 M-tiles x 4 N-tiles = 8 v8f accumulators.
//    B chunks (32 rows x 64 cols) DOUBLE-BUFFERED in LDS, one barrier per
//    chunk; next chunk's global loads issued before the current chunk's
//    64 WMMAs so L2 latency is hidden. LDS is row-pair-interleaved:
//        lds[(pair*64 + col)*2 + (row&1)]
//    so each B fragment is ONE aligned ds_load_b64 (no v_mov repacking).
//    V_WMMA_F32_16X16X4_F32 fragment layouts per ISA 7.12.2:
//      A 16x4 : lanes 0-15 M=lane {K,K+1}; lanes 16-31 same M, {K+2,K+3}
//      B 4x16 : v0 = rows {K, K+2}, v1 = rows {K+1, K+3} (lane halves)
//      C 16x16: vj -> (M=j | j+8, N=lane&15)
// ---------------------------------------------------------------------------
__global__ __launch_bounds__(256) void k_gemm(const float* __restrict__ Wm,
                                              const float* __restrict__ B,
                                              const float* __restrict__ Add,
                                              float* __restrict__ Out) {
    __shared__ float ldsB[2][32 * 64];   // 2 x 8 KB, pair-interleaved

    const int tid  = threadIdx.x;
    const int wave = tid >> 5;
    const int lane = tid & 31;
    const int hh   = lane >> 4;   // lane half (0/1)
    const int l16  = lane & 15;

    const int mrow = wave * 32 + l16;     // A row for M-tile 0 (tile 1: +16)
    const int n0   = blockIdx.x * 64;

    // staging decomposition: thread -> row pair sp (16), col group sc (16x4)
    const int sp = tid >> 4;              // 0..15  (rows 2sp, 2sp+1)
    const int sc = (tid & 15) * 4;        // 0..60

    const float* __restrict__ Bp0 = &B[(size_t)(2 * sp)     * NN + n0 + sc];
    const float* __restrict__ Bp1 = &B[(size_t)(2 * sp + 1) * NN + n0 + sc];

    v8f acc[8];
    #pragma unroll
    for (int i = 0; i < 8; ++i) acc[i] = v8f{};

    // prologue: stage chunk 0 into buffer 0
    {
        float4 r0 = *(const float4*)(Bp0);
        float4 r1 = *(const float4*)(Bp1);
        #pragma unroll
        for (int i = 0; i < 4; ++i) {
            v2f pr; pr.x = ((const float*)&r0)[i]; pr.y = ((const float*)&r1)[i];
            *(v2f*)&ldsB[0][(sp * 64 + sc + i) * 2] = pr;
        }
    }

    #pragma unroll
    for (int kc = 0; kc < KK; kc += 32) {
        const int buf  = (kc >> 5) & 1;
        const bool more = (kc + 32 < KK);

        __syncthreads();   // buf staged; alternate buffer free for writing

        // prefetch next chunk into registers (overlaps with WMMAs below)
        float4 p0, p1;
        if (more) {
            p0 = *(const float4*)(Bp0 + (size_t)(kc + 32) * NN);
            p1 = *(const float4*)(Bp1 + (size_t)(kc + 32) * NN);
        }

        #pragma unroll
        for (int k = 0; k < 32; k += 4) {
            // A fragments: rows mrow / mrow+16, cols kc+k+2*hh, +1 (8B pair)
            v2f a0 = *(const v2f*)&Wm[(mrow)      * KK + kc + k + 2 * hh];
            v2f a1 = *(const v2f*)&Wm[(mrow + 16) * KK + kc + k + 2 * hh];
            const int pbase = (k >> 1) + hh;     // row pair (k+2*hh, k+2*hh+1)
            #pragma unroll
            for (int t = 0; t < 4; ++t) {
                v2f b = *(const v2f*)&ldsB[buf][(pbase * 64 + t * 16 + l16) * 2];
                acc[t] = __builtin_amdgcn_wmma_f32_16x16x4_f32(
                    false, a0, false, b, (short)0, acc[t], false, false);
                acc[4 + t] = __builtin_amdgcn_wmma_f32_16x16x4_f32(
                    false, a1, false, b, (short)0, acc[4 + t], false, false);
            }
        }

        // write next chunk into the alternate buffer (readers of it finished
        // before this iteration's top barrier)
        if (more) {
            #pragma unroll
            for (int i = 0; i < 4; ++i) {
                v2f pr; pr.x = ((const float*)&p0)[i]; pr.y = ((const float*)&p1)[i];
                *(v2f*)&ldsB[buf ^ 1][(sp * 64 + sc + i) * 2] = pr;
            }
        }
    }

    // ---- epilogue: D layout -> global (+ optional addend) ----
    #pragma unroll
    for (int mt = 0; mt < 2; ++mt) {
        #pragma unroll
        for (int t = 0; t < 4; ++t) {
            #pragma unroll
            for (int j = 0; j < 8; ++j) {
                int m = wave * 32 + mt * 16 + j + 8 * hh;
                int n = n0 + t * 16 + l16;
                float v = acc[mt * 4 + t][j];
                if (Add) v += Add[(size_t)m * NN + n];
                Out[(size_t)m * NN + n] = v;
            }
        }
    }
}

// ---------------------------------------------------------------------------
// 5) shrinkage / dual update (float4 vectorized):
//    v = x+u; vp = max(v,0); z = max(vp-kappa,0); u' = u+x-z; T = z-u'
// ---------------------------------------------------------------------------
__global__ __launch_bounds__(256) void k_elem(const float* __restrict__ X,
                                              float* __restrict__ U,
                                              float* __restrict__ T) {
    int i = (blockIdx.x * 256 + threadIdx.x) * 4;
    float4 x = *(const float4*)&X[i];
    float4 u = *(const float4*)&U[i];
    float4 un, tt;
    #pragma unroll
    for (int c = 0; c < 4; ++c) {
        float xc = ((const float*)&x)[c];
        float uc = ((const float*)&u)[c];
        float vp = fmaxf(xc + uc, 0.0f);
        float z  = fmaxf(vp - KAPPA_C, 0.0f);
        float uu = uc + xc - z;
        ((float*)&un)[c] = uu;
        ((float*)&tt)[c] = z - uu;
    }
    *(float4*)&U[i] = un;
    *(float4*)&T[i] = tt;
}

// ---------------------------------------------------------------------------
// 6) out[n][k] = X[k][n]   (LDS-tiled 32x32 transpose)
// ---------------------------------------------------------------------------
__global__ __launch_bounds__(256) void k_transpose(const float* __restrict__ X,
                                                   float* __restrict__ Out) {
    __shared__ float tile[32][33];
    int bx = blockIdx.x;          // n tile (256)
    int by = blockIdx.y;          // k tile (8)
    int tx = threadIdx.x;         // 32
    int ty = threadIdx.y;         // 8
    int n = bx * 32 + tx;
    #pragma unroll
    for (int j = ty; j < 32; j += 8)
        tile[j][tx] = X[(size_t)(by * 32 + j) * NN + n];
    __syncthreads();
    int k = by * 32 + tx;
    #pragma unroll
    for (int j = ty; j < 32; j += 8)
        Out[(size_t)(bx * 32 + j) * KK + k] = tile[tx][j];
}

// ---------------------------------------------------------------------------
extern "C" void kernel_launch(void* const* d_in, const int* in_sizes, int n_in,
                              void* d_out, int out_size, void* d_ws, size_t ws_size,
                              hipStream_t stream) {
    (void)in_sizes; (void)n_in; (void)out_size; (void)ws_size;
    const float* A    = (const float*)d_in[0];   // [256, 256]
    const float* data = (const float*)d_in[1];   // [8192, 256]
    float* out = (float*)d_out;                  // [8192, 256]

    float* ws  = (float*)d_ws;
    float* Aug = ws;                     // 256*512
    float* Wm  = Aug + 256 * 512;        // 256*256
    float* AtY = Wm  + 256 * 256;        // 256*8192
    float* C0  = AtY + (size_t)KK * NN;
    float* X   = C0  + (size_t)KK * NN;
    float* U   = X   + (size_t)KK * NN;
    float* T   = U   + (size_t)KK * NN;

    // U = T = 0  (contiguous)
    hipMemsetAsync(U, 0, 2 * (size_t)KK * NN * sizeof(float), stream);

    // setup: Gram + identity, invert, AtY, C0 = W @ AtY
    k_gram  <<<512, 256, 0, stream>>>(A, Aug);
    k_invert<<<1, 1024, 0, stream>>>(Aug, Wm);
    k_atY   <<<dim3(32, 256), 256, 0, stream>>>(A, data, AtY);
    k_gemm  <<<dim3(NN / 64, 1), 256, 0, stream>>>(Wm, AtY, nullptr, C0);

    // ADMM main loop: X = W @ T + C0 ; elementwise shrinkage
    for (int it = 0; it < NITER; ++it) {
        k_gemm<<<dim3(NN / 64, 1), 256, 0, stream>>>(Wm, T, C0, X);
        k_elem<<<(KK * NN) / (256 * 4), 256, 0, stream>>>(X, U, T);
    }

    // out = X^T
    k_transpose<<<dim3(NN / 32, KK / 32), dim3(32, 8), 0, stream>>>(X, out);
}